// GATConvBlock_72164040507950
// MI455X (gfx1250) — compile-verified
//
#include <hip/hip_runtime.h>
#include <math.h>

#define N_NODES 100000
#define N_EDGES 1600000
#define D_IN 64
#define D_OUT 64
#define D_EDGE 16
#define NEG_SLOPE 0.01f
#define LN_EPS 1e-5f

typedef __attribute__((ext_vector_type(2))) float v2f;
typedef __attribute__((ext_vector_type(8))) float v8f;

// float atomic max via signed-int max (>=0) / unsigned-int min (<0); requires -inf init.
__device__ __forceinline__ void atomicMaxFloat(float* addr, float v) {
    if (v >= 0.0f) {
        atomicMax((int*)addr, __float_as_int(v));
    } else {
        atomicMin((unsigned int*)addr, __float_as_uint(v));
    }
}

// ---------------------------------------------------------------------------
// Kernel 1: init h (=d_out) to 0, mmax to -inf, denom to 0, and compute
// w[j] = sum_k att_l[k] * W1[k][j]  (j < 80)
// ---------------------------------------------------------------------------
__global__ void init_kernel(float* __restrict__ h, float* __restrict__ mmax,
                            float* __restrict__ denom, float* __restrict__ w,
                            const float* __restrict__ att_l, const float* __restrict__ W1) {
    int i = blockIdx.x * blockDim.x + threadIdx.x;
    if (i < N_NODES * D_OUT) h[i] = 0.0f;
    if (i < N_NODES) { mmax[i] = -__builtin_inff(); denom[i] = 0.0f; }
    if (i < D_IN + D_EDGE) {
        float s = 0.0f;
        #pragma unroll
        for (int k = 0; k < D_OUT; ++k) s += att_l[k] * W1[k * (D_IN + D_EDGE) + i];
        w[i] = s;
    }
}

// ---------------------------------------------------------------------------
// Kernel 2: x_t = x @ W2^T via V_WMMA_F32_16X16X4_F32, plus per-node scalars
//   ar[n] = x[n] . att_r    al[n] = x[n] . w[:64]
// One wave per 16-row tile; 8 waves / block; W2 staged in LDS.
// A 16x4 f32 layout: lanes 0-15 -> K=0(v0),K=1(v1); lanes 16-31 -> K=2,K=3.
// B 4x16 layout mirrors: v0 holds K=0 (lanes 0-15) / K=2 (lanes 16-31), v1 K=1/K=3.
// D 16x16 f32: lane<16 -> N=lane, M=vgpr; lane>=16 -> N=lane-16, M=vgpr+8.
// ---------------------------------------------------------------------------
__global__ void __launch_bounds__(256)
node_gemm_kernel(const float* __restrict__ x, const float* __restrict__ W2,
                 const float* __restrict__ att_r, const float* __restrict__ w,
                 float* __restrict__ xt, float* __restrict__ ar, float* __restrict__ al) {
    __shared__ float sW2[D_OUT * D_IN];   // row-major W2[k][j]
    __shared__ float sAr[D_IN];
    __shared__ float sWx[D_IN];

    int tid = threadIdx.x;
    for (int t = tid; t < D_OUT * D_IN; t += 256) sW2[t] = W2[t];
    if (tid < D_IN) { sAr[tid] = att_r[tid]; sWx[tid] = w[tid]; }
    __syncthreads();

    int wave = tid >> 5;
    int lane = tid & 31;
    int tile = blockIdx.x * 8 + wave;             // 16-row tile index (uniform per wave)
    if (tile >= N_NODES / 16) return;             // 100000 % 16 == 0: tiles always full
    int r0   = tile * 16;
    int half = lane >> 4;                         // 0: K lo pair, 1: K hi pair
    int m    = lane & 15;                         // row (A) / col (B,D) within tile

    v8f acc0 = {}, acc1 = {}, acc2 = {}, acc3 = {};
    float s_ar = 0.0f, s_al = 0.0f;

    const float* xrow = x + (size_t)(r0 + m) * D_IN;
    #pragma unroll
    for (int kb = 0; kb < D_IN; kb += 4) {
        int c0 = kb + (half ? 2 : 0);
        float a0 = xrow[c0];
        float a1 = xrow[c0 + 1];
        v2f a; a.x = a0; a.y = a1;
        s_ar += a0 * sAr[c0] + a1 * sAr[c0 + 1];
        s_al += a0 * sWx[c0] + a1 * sWx[c0 + 1];

        v2f b;
        // N-tile 0 (out cols 0..15):  B[k][n] = W2[n][k]
        b.x = sW2[(0  + m) * D_IN + c0]; b.y = sW2[(0  + m) * D_IN + c0 + 1];
        acc0 = __builtin_amdgcn_wmma_f32_16x16x4_f32(false, a, false, b, (short)0, acc0, false, false);
        // N-tile 1
        b.x = sW2[(16 + m) * D_IN + c0]; b.y = sW2[(16 + m) * D_IN + c0 + 1];
        acc1 = __builtin_amdgcn_wmma_f32_16x16x4_f32(false, a, false, b, (short)0, acc1, false, false);
        // N-tile 2
        b.x = sW2[(32 + m) * D_IN + c0]; b.y = sW2[(32 + m) * D_IN + c0 + 1];
        acc2 = __builtin_amdgcn_wmma_f32_16x16x4_f32(false, a, false, b, (short)0, acc2, false, false);
        // N-tile 3
        b.x = sW2[(48 + m) * D_IN + c0]; b.y = sW2[(48 + m) * D_IN + c0 + 1];
        acc3 = __builtin_amdgcn_wmma_f32_16x16x4_f32(false, a, false, b, (short)0, acc3, false, false);
    }

    // combine lane L (K = 0,1 mod 4) with lane L+16 (K = 2,3 mod 4)
    s_ar += __shfl_xor(s_ar, 16, 32);
    s_al += __shfl_xor(s_al, 16, 32);
    if (half == 0) { ar[r0 + m] = s_ar; al[r0 + m] = s_al; }

    #pragma unroll
    for (int vr = 0; vr < 8; ++vr) {
        int M = vr + 8 * half;
        float* orow = xt + (size_t)(r0 + M) * D_OUT;
        orow[0  + m] = acc0[vr];
        orow[16 + m] = acc1[vr];
        orow[32 + m] = acc2[vr];
        orow[48 + m] = acc3[vr];
    }
}

// ---------------------------------------------------------------------------
// Kernel 3: per-edge score e = leaky_relu(ar[dst] + al[src] + edge_attr . w_e),
// store e, atomic-max into mmax[dst]. One thread per edge.
// ---------------------------------------------------------------------------
__global__ void __launch_bounds__(256)
edge_pass1_kernel(const float* __restrict__ edge_attr,
                  const int* __restrict__ src, const int* __restrict__ dst,
                  const float* __restrict__ ar, const float* __restrict__ al,
                  const float* __restrict__ w,
                  float* __restrict__ e_buf, float* __restrict__ mmax) {
    int i = blockIdx.x * blockDim.x + threadIdx.x;
    if (i >= N_EDGES) return;
    int s = src[i], d = dst[i];
    float sum = ar[d] + al[s];
    const float4* ea = (const float4*)(edge_attr + (size_t)i * D_EDGE);
    #pragma unroll
    for (int q = 0; q < 4; ++q) {
        float4 v = ea[q];
        sum += v.x * w[D_IN + 4 * q]     + v.y * w[D_IN + 4 * q + 1]
             + v.z * w[D_IN + 4 * q + 2] + v.w * w[D_IN + 4 * q + 3];
    }
    float e = sum > 0.0f ? sum : NEG_SLOPE * sum;
    e_buf[i] = e;
    atomicMaxFloat(&mmax[d], e);
}

// ---------------------------------------------------------------------------
// Kernel 4: one wave per edge. ex = exp(e - mmax[dst]); atomicAdd denom[dst];
// h[dst,:] += ex * x_t[src,:]  (normalization by denom deferred to finish).
// Lane j handles columns 2j, 2j+1 (float2).
// ---------------------------------------------------------------------------
__global__ void __launch_bounds__(256)
edge_pass2_kernel(const int* __restrict__ src, const int* __restrict__ dst,
                  const float* __restrict__ e_buf, const float* __restrict__ mmax,
                  const float* __restrict__ xt,
                  float* __restrict__ denom, float* __restrict__ h) {
    int widx = (blockIdx.x * blockDim.x + threadIdx.x) >> 5;
    int lane = threadIdx.x & 31;
    if (widx >= N_EDGES) return;
    int s = src[widx], d = dst[widx];
    float ex = __expf(e_buf[widx] - mmax[d]);
    if (lane == 0) atomicAdd(&denom[d], ex);
    float2 v = ((const float2*)(xt + (size_t)s * D_OUT))[lane];
    float* hp = h + (size_t)d * D_OUT + 2 * lane;
    atomicAdd(hp,     v.x * ex);
    atomicAdd(hp + 1, v.y * ex);
}

// ---------------------------------------------------------------------------
// Kernel 5: one wave per node. out = LN(h/denom + bias) * gamma + beta.
// In-place on d_out (h aliases out).
// ---------------------------------------------------------------------------
__global__ void __launch_bounds__(256)
node_finish_kernel(float* __restrict__ h, const float* __restrict__ denom,
                   const float* __restrict__ bias, const float* __restrict__ gamma,
                   const float* __restrict__ beta) {
    int n    = (blockIdx.x * blockDim.x + threadIdx.x) >> 5;
    int lane = threadIdx.x & 31;
    if (n >= N_NODES) return;
    float den   = denom[n];
    float scale = den > 0.0f ? 1.0f / den : 0.0f;   // isolated nodes -> h = 0
    float2 v = ((const float2*)(h + (size_t)n * D_OUT))[lane];
    float2 b = ((const float2*)bias)[lane];
    float o0 = v.x * scale + b.x;
    float o1 = v.y * scale + b.y;

    float sum = o0 + o1;
    #pragma unroll
    for (int off = 16; off > 0; off >>= 1) sum += __shfl_xor(sum, off, 32);
    float mu = sum * (1.0f / 64.0f);
    float d0 = o0 - mu, d1 = o1 - mu;
    float vs = d0 * d0 + d1 * d1;
    #pragma unroll
    for (int off = 16; off > 0; off >>= 1) vs += __shfl_xor(vs, off, 32);
    float rs = rsqrtf(vs * (1.0f / 64.0f) + LN_EPS);

    float2 g  = ((const float2*)gamma)[lane];
    float2 be = ((const float2*)beta)[lane];
    float2 r;
    r.x = d0 * rs * g.x + be.x;
    r.y = d1 * rs * g.y + be.y;
    ((float2*)(h + (size_t)n * D_OUT))[lane] = r;
}

extern "C" void kernel_launch(void* const* d_in, const int* in_sizes, int n_in,
                              void* d_out, int out_size, void* d_ws, size_t ws_size,
                              hipStream_t stream) {
    (void)in_sizes; (void)n_in; (void)out_size; (void)ws_size;
    const float* x         = (const float*)d_in[0];
    const float* edge_attr = (const float*)d_in[1];
    const float* W1        = (const float*)d_in[2];
    const float* W2        = (const float*)d_in[3];
    const float* att_l     = (const float*)d_in[4];
    const float* att_r     = (const float*)d_in[5];
    const float* bias      = (const float*)d_in[6];
    const float* gamma     = (const float*)d_in[7];
    const float* beta      = (const float*)d_in[8];
    const int*   src       = (const int*)d_in[9];
    const int*   dst       = (const int*)d_in[10];

    float* out = (float*)d_out;          // doubles as the h accumulator
    float* ws  = (float*)d_ws;
    float* xt    = ws;                                   // N*64
    float* ar    = xt + (size_t)N_NODES * D_OUT;         // N
    float* al    = ar + N_NODES;                         // N
    float* mmax  = al + N_NODES;                         // N
    float* denom = mmax + N_NODES;                       // N
    float* w     = denom + N_NODES;                      // 128 (80 used)
    float* e_buf = w + 128;                              // E

    init_kernel<<<(N_NODES * D_OUT + 255) / 256, 256, 0, stream>>>(
        out, mmax, denom, w, att_l, W1);

    node_gemm_kernel<<<(N_NODES / 16 + 7) / 8, 256, 0, stream>>>(
        x, W2, att_r, w, xt, ar, al);

    edge_pass1_kernel<<<(N_EDGES + 255) / 256, 256, 0, stream>>>(
        edge_attr, src, dst, ar, al, w, e_buf, mmax);

    edge_pass2_kernel<<<(N_EDGES + 7) / 8, 256, 0, stream>>>(
        src, dst, e_buf, mmax, xt, denom, out);

    node_finish_kernel<<<(N_NODES + 7) / 8, 256, 0, stream>>>(
        out, denom, bias, gamma, beta);
}